// SPUPointwise_60352880444029
// MI455X (gfx1250) — compile-verified
//
#include <hip/hip_runtime.h>

// Pointwise SPU linear-relaxation bounds (DeepPoly-style) for MI455X/gfx1250.
//
// Roofline: 16 B/elem * 4.19M elem = 67 MB @ 23.3 TB/s => ~2.9 us floor.
// Strictly elementwise => WMMA not applicable. The path used here:
//   - b128 non-temporal loads/stores (global_load_b128 th:TH_LOAD_NT)
//   - hardware v_exp_f32 / v_rcp_f32 instead of libm exp/cosh/div
//   - algebraic reduction: sigmoid identity replaces cosh, closed forms for
//     the x>=0 tangents, so ~3 exp + ~5 rcp per element
//   - dual-issue/packed VALU (VOPD, v_pk_fma_f32) via straight-line code
// Tail handled by a separate micro-kernel so the hot kernel stays tight.

typedef __attribute__((ext_vector_type(4))) float v4f;

__device__ __forceinline__ float fast_rcp(float x) {
    return __builtin_amdgcn_rcpf(x);   // v_rcp_f32, ~1 ulp
}

// SPU(x) = x^2 - 0.5            (x >= 0)
//        = sigmoid(-x) - 1      (x <  0)   ; sigmoid(-x) = 1/(1+e^x)
__device__ __forceinline__ float spu_f(float x) {
    float e   = __expf(x);                    // v_exp_f32
    float s   = fast_rcp(1.0f + e);           // sigmoid(-x)
    float neg = s - 1.0f;
    float pos = x * x - 0.5f;
    return (x >= 0.0f) ? pos : neg;
}

// Per-element output pair (nl, nu), following the reference dataflow with
// exact algebraic strength reduction (see comments).
__device__ __forceinline__ void spu_bounds(float l, float u, float& out_nl, float& out_nu) {
    float sl = spu_f(l);
    float su = spu_f(u);

    // chord over [l, u]
    float a_lu = (su - sl) * fast_rcp(u - l + 1e-8f);
    float b_lu = sl - l * a_lu;

    // tangent at mid = (l+u)/2.
    //  mid >= 0 : slope 2*mid, intercept spu(mid) - 2*mid*mid
    //  mid <  0 : with s = sigmoid(-mid) = 1/(1+e^mid):
    //             spu(mid) = s - 1,  spu_grad(mid) = -0.5/(cosh(mid)+1) = -s*(1-s)
    //             (exact identity; clamp: grad = 0 for mid <= -50)
    float mid = 0.5f * (l + u);
    float a_t, b_t;
    {
        float e  = __expf(mid);
        float s  = fast_rcp(1.0f + e);
        float gn = (mid <= -50.0f) ? 0.0f : -(s - s * s);
        float an = gn;
        float bn = (s - 1.0f) - gn * mid;
        float ap = 2.0f * mid;
        float bp = (mid * mid - 0.5f) - ap * mid;
        a_t = (mid >= 0.0f) ? ap : an;
        b_t = (mid >= 0.0f) ? bp : bn;
    }

    bool neg    = (u < 0.0f);
    bool pos    = (!neg) && (l > 0.0f);
    bool mixed  = (!neg) && (!pos);
    bool normal = mixed && (su > sl);

    // mixed lower-bound candidates
    // c1: chord from (l, sl) to (0, -0.5)
    float a1 = (-0.5f - sl) * fast_rcp(-l + 1e-8f);
    float b1 = sl - l * a1;
    // c2: tangent at h = u/2. Only consumed when `normal` (=> u >= 0 => h >= 0),
    // so grad(h) = 2h and spu(h) = h*h - 0.5 exactly — no transcendental needed.
    float h  = 0.5f * u;
    float a2 = 2.0f * h;
    float b2 = (h * h - 0.5f) - a2 * h;

    // area proxy: -(u-l)*(a*(l+u) + 2b)
    float span = u - l;
    float lpu  = l + u;
    float ar0  = span;                                 // -(u-l)*(0 + 2*(-0.5))
    float ar1  = -span * (a1 * lpu + 2.0f * b1);
    float ar2  = normal ? (-span * (a2 * lpu + 2.0f * b2)) : __builtin_inff();

    // argmin with first-occurrence tie-breaking (matches jnp.argmin)
    int   idx  = 0;
    float best = ar0;
    if (ar1 < best) { best = ar1; idx = 1; }
    if (ar2 < best) { idx = 2; }
    float a_m = (idx == 0) ? 0.0f  : ((idx == 1) ? a1 : a2);
    float b_m = (idx == 0) ? -0.5f : ((idx == 1) ? b1 : b2);

    float lw = neg ? a_lu : (pos ? a_t : a_m);
    float lb = neg ? b_lu : (pos ? b_t : b_m);
    bool  pn = pos || normal;
    float uw = neg ? a_t : (pn ? a_lu : 0.0f);
    float ub = neg ? b_t : (pn ? b_lu : sl);

    // backsubstitution against constant input bounds
    out_nl = lb + ((lw > 0.0f) ? lw * l : lw * u);
    out_nu = ub + ((uw > 0.0f) ? uw * u : uw * l);
}

// Hot kernel: pure 128-bit vector streaming, no tail code inlined.
__global__ __launch_bounds__(256) void spu_relax_vec_kernel(
    const float* __restrict__ l, const float* __restrict__ u,
    float* __restrict__ nl, float* __restrict__ nu, int n4) {
    int i = blockIdx.x * blockDim.x + threadIdx.x;
    if (i >= n4) return;

    v4f lv = __builtin_nontemporal_load(reinterpret_cast<const v4f*>(l) + i);
    v4f uv = __builtin_nontemporal_load(reinterpret_cast<const v4f*>(u) + i);
    v4f nlv, nuv;
#pragma unroll
    for (int k = 0; k < 4; ++k) {
        float a, b;
        spu_bounds(lv[k], uv[k], a, b);
        nlv[k] = a;
        nuv[k] = b;
    }
    __builtin_nontemporal_store(nlv, reinterpret_cast<v4f*>(nl) + i);
    __builtin_nontemporal_store(nuv, reinterpret_cast<v4f*>(nu) + i);
}

// Tail kernel: at most 3 elements; launched only when n % 4 != 0.
__global__ __launch_bounds__(32) void spu_relax_tail_kernel(
    const float* __restrict__ l, const float* __restrict__ u,
    float* __restrict__ nl, float* __restrict__ nu, int start, int n) {
    int t = start + (int)threadIdx.x;
    if (t >= n) return;
    float a, b;
    spu_bounds(l[t], u[t], a, b);
    nl[t] = a;
    nu[t] = b;
}

extern "C" void kernel_launch(void* const* d_in, const int* in_sizes, int n_in,
                              void* d_out, int out_size, void* d_ws, size_t ws_size,
                              hipStream_t stream) {
    const float* l = static_cast<const float*>(d_in[0]);
    const float* u = static_cast<const float*>(d_in[1]);
    float* out = static_cast<float*>(d_out);

    const int n  = in_sizes[0];        // 64 * 65536 = 4,194,304
    float* nl = out;                   // first output, flat
    float* nu = out + n;               // second output, flat

    const int n4    = n / 4;
    const int block = 256;             // 8 wave32 waves per block

    if (n4 > 0) {
        int grid = (n4 + block - 1) / block;
        spu_relax_vec_kernel<<<grid, block, 0, stream>>>(l, u, nl, nu, n4);
    }
    // Deterministic: launch decision depends only on input sizes.
    if (n - 4 * n4 > 0) {
        spu_relax_tail_kernel<<<1, 32, 0, stream>>>(l, u, nl, nu, 4 * n4, n);
    }
}